// LieGroupMulti_18751827214620
// MI455X (gfx1250) — compile-verified
//
#include <hip/hip_runtime.h>

// Fused Lie-algebra einsum + batched 10x10 matrix exponential for MI455X (gfx1250).
//
//   lie_alg[b,w,:,:] = sum_z z[b,w,z] * basis[w,z,:,:]      (fp32 WMMA 16x16x4)
//   out[b,w,:,:]     = expm(lie_alg[b,w,:,:])               (scaling&squaring + Taylor-8)
//
// Roofline: ~164 MB of fp32 traffic @ 23.3 TB/s => ~7us floor; 6.7 GFLOP of
// contraction is far below the matrix-op ceiling, so precision stays fp32
// end-to-end via V_WMMA_F32_16X16X4_F32. The z stream (134 MB) is staged
// through LDS with coalesced GLOBAL_LOAD_ASYNC_TO_LDS_B128 (ASYNCcnt),
// basis (3.3 MB) stays L2-resident via direct broadcast loads.

typedef __attribute__((ext_vector_type(2))) float v2f;
typedef __attribute__((ext_vector_type(8))) float v8f;

#define B_TOT   4096
#define NUM_WS  16
#define Z_DIM   512
#define MAT     10
#define MM      (MAT * MAT)   // 100
#define NPAD    112           // 7 WMMA n-tiles of 16 cover the 100 ij columns
#define NT      7
#define RPITCH  11            // LDS row pitch inside a 10x10 matrix (bank-friendly)
#define MPITCH  112           // LDS pitch per matrix
#define THREADS 128
#define WAVES   4
#define KSLICE  (Z_DIM / WAVES)   // 128
#define ZPITCH  516           // z-tile LDS row pitch (dwords): banks 4r+k, conflict-free
#define SBIG    (16 * ZPITCH) // 8256 dwords: z staging, later aliased by wave partials

__device__ __forceinline__ void async_load_b128_to_lds(void* lds_dst, const void* gsrc) {
    // GLOBAL_LOAD_ASYNC_TO_LDS_B128: per-lane 16B memory->LDS, tracked by ASYNCcnt.
    // Low 32 bits of a generic pointer into LDS aperture == LDS byte address.
    const unsigned int       ldsa = (unsigned int)(unsigned long long)(uintptr_t)lds_dst;
    const unsigned long long gva  = (unsigned long long)(uintptr_t)gsrc;
    asm volatile("global_load_async_to_lds_b128 %0, %1, off"
                 :: "v"(ldsa), "v"(gva) : "memory");
}

__device__ __forceinline__ void wait_asynccnt0() {
#if __has_builtin(__builtin_amdgcn_s_wait_asynccnt)
    __builtin_amdgcn_s_wait_asynccnt(0);
#else
    asm volatile("s_wait_asynccnt 0" ::: "memory");
#endif
}

__global__ __launch_bounds__(THREADS)
void lie_expm_fused(const float* __restrict__ z,
                    const float* __restrict__ basis,
                    float* __restrict__ out)
{
    __shared__ float sBig[SBIG];       // phase 0/1: z tile; phase 2: wave partials
    __shared__ float sA[16 * MPITCH];  // lie_alg matrices (row pitch 11)
    __shared__ float sG[16 * MPITCH];  // squaring workspace
    __shared__ float sRed[16 * 16];    // row-sum reduction scratch
    __shared__ int   sS[16];           // per-matrix squaring count

    const int tid  = threadIdx.x;
    const int wave = tid >> 5;
    const int lane = tid & 31;
    const int l16  = lane & 15;
    const int lhi  = lane >> 4;        // 0 or 1 (half-wave)

    const int b0 = blockIdx.x * 16;    // 16 batch rows per block
    const int w  = blockIdx.y;

    // ---------------- Phase 0: coalesced async stage of the 16x512 z tile -----
    // Row i of the tile = z[b0+i, w, :]. Thread t loads 16B at k=4t of row i:
    // 128 threads * 16B cover one 2KB row per iteration, fully coalesced.
    {
        const int k4 = 4 * tid;
#pragma unroll
        for (int i = 0; i < 16; ++i) {
            const float* gp = z + (((size_t)(b0 + i) * NUM_WS) + w) * Z_DIM + k4;
            async_load_b128_to_lds(&sBig[i * ZPITCH + k4], gp);
        }
    }
    wait_asynccnt0();
    __syncthreads();

    // ---------------- Phase 1: C = Ztile * Basis via V_WMMA_F32_16X16X4_F32 ----
    // A (16x4 f32): lane m=l16 holds K=k0+2*lhi (+1): ds_load_b64 from sBig,
    //               conflict-free (addr dword = 516*m + k, 4m+k distinct mod 64).
    // B (4x16 f32): lane n=l16 holds rows K=k0+2*lhi (+1) of basis (L2 broadcast).
    const float* bw = basis + (size_t)w * Z_DIM * MM;

    // tail n-tile (nt=6): clamp column + 0/1 mask, hoisted out of the K loop
    const int   n6  = 96 + l16;
    const int   nc6 = (n6 < MM) ? n6 : 0;
    const float m6  = (n6 < MM) ? 1.0f : 0.0f;

    v8f acc[NT];
#pragma unroll
    for (int nt = 0; nt < NT; ++nt)
#pragma unroll
        for (int e = 0; e < 8; ++e) acc[nt][e] = 0.0f;

    const int kbeg = wave * KSLICE;    // waves split K=512 into 4 slices
    const float* sZrow = &sBig[l16 * ZPITCH];
    for (int k0 = kbeg; k0 < kbeg + KSLICE; k0 += 4) {
        const int ka = k0 + 2 * lhi;
        v2f a;
        a.x = sZrow[ka];
        a.y = sZrow[ka + 1];
        const float* bk = bw + (size_t)ka * MM;   // row ka; row ka+1 at +MM
#pragma unroll
        for (int nt = 0; nt < 6; ++nt) {
            v2f b;
            b.x = bk[nt * 16 + l16];
            b.y = bk[MM + nt * 16 + l16];
            acc[nt] = __builtin_amdgcn_wmma_f32_16x16x4_f32(
                false, a, false, b, (short)0, acc[nt], false, false);
        }
        v2f b6;
        b6.x = bk[nc6] * m6;
        b6.y = bk[MM + nc6] * m6;
        acc[6] = __builtin_amdgcn_wmma_f32_16x16x4_f32(
            false, a, false, b6, (short)0, acc[6], false, false);
    }

    __syncthreads();                   // all waves done reading the z tile

    // ---------------- Phase 2: wave partials (alias sBig) + fixed-order reduce -
    // D layout: VGPR v -> row m = v + 8*lhi, col = l16 (per ISA 16x16 f32 C/D map)
    float* sPart = sBig;               // [WAVES][16*NPAD] = 7168 <= SBIG dwords
#pragma unroll
    for (int nt = 0; nt < NT; ++nt)
#pragma unroll
        for (int v = 0; v < 8; ++v)
            sPart[wave * (16 * NPAD) + (v + 8 * lhi) * NPAD + nt * 16 + l16] = acc[nt][v];
    __syncthreads();

    for (int idx = tid; idx < 16 * NPAD; idx += THREADS) {
        float s = sPart[0 * (16 * NPAD) + idx] + sPart[1 * (16 * NPAD) + idx]
                + sPart[2 * (16 * NPAD) + idx] + sPart[3 * (16 * NPAD) + idx];
        const int gm = idx / NPAD;
        const int n  = idx % NPAD;
        if (n < MM) {
            const int i = n / MAT, j = n % MAT;
            sA[gm * MPITCH + i * RPITCH + j] = s;
        }
    }
    __syncthreads();

    // ---------------- Phase 3: expm per matrix (16 lanes / matrix) -------------
    // All cross-lane traffic below is intra-wave LDS: wave32 DS ops are in
    // program order per wave, so no workgroup barriers are needed (and none are
    // legal inside the divergent squaring loop).
    const int grp = tid >> 4;          // 8 groups of 16 lanes, 2 matrices each
    const int l   = tid & 15;
    for (int m = grp; m < 16; m += 8) {
        float* A = &sA[m * MPITCH];
        float* G = &sG[m * MPITCH];
        const int  r      = l;
        const bool active = (r < MAT);

        // inf-norm -> squaring count s so that ||A||/2^s <= 0.25
        float rs = 0.0f;
        if (active)
            for (int j = 0; j < MAT; ++j) rs += fabsf(A[r * RPITCH + j]);
        sRed[m * 16 + l] = active ? rs : 0.0f;
        if (l == 0) {
            float nrm = 0.0f;
            for (int t = 0; t < MAT; ++t) nrm = fmaxf(nrm, sRed[m * 16 + t]);
            int s = 0;
            while (nrm > 0.25f && s < 30) { nrm *= 0.5f; ++s; }
            sS[m] = s;
        }
        const int s = sS[m];
        float scale = 1.0f;
        for (int t = 0; t < s; ++t) scale *= 0.5f;

        // X = A * 2^-s (in place); P = X (row in regs); R = I + X (row in regs)
        float P[MAT], R[MAT];
        if (active) {
            for (int j = 0; j < MAT; ++j) {
                const float x = A[r * RPITCH + j] * scale;
                A[r * RPITCH + j] = x;
                P[j] = x;
                R[j] = x + (r == j ? 1.0f : 0.0f);
            }
        }

        // Taylor to order 8: ||X|| <= 0.25 -> truncation error < 1e-10 rel
        for (int k = 2; k <= 8; ++k) {
            const float inv = 1.0f / (float)k;
            if (active) {
                float nP[MAT];
                for (int j = 0; j < MAT; ++j) {
                    float a2 = 0.0f;
                    for (int kk = 0; kk < MAT; ++kk)
                        a2 += P[kk] * A[kk * RPITCH + j];  // broadcast LDS read
                    nP[j] = a2 * inv;
                }
                for (int j = 0; j < MAT; ++j) { P[j] = nP[j]; R[j] += nP[j]; }
            }
        }

        // squaring: G <- G*G, s times (row of G kept in regs as R)
        if (active)
            for (int j = 0; j < MAT; ++j) G[r * RPITCH + j] = R[j];
        for (int q = 0; q < s; ++q) {
            if (active) {
                float nR[MAT];
                for (int j = 0; j < MAT; ++j) {
                    float a2 = 0.0f;
                    for (int kk = 0; kk < MAT; ++kk)
                        a2 += R[kk] * G[kk * RPITCH + j];
                    nR[j] = a2;
                }
                for (int j = 0; j < MAT; ++j) { R[j] = nR[j]; G[r * RPITCH + j] = nR[j]; }
            }
        }

        if (active) {
            float* o = out + (((size_t)(b0 + m) * NUM_WS + w) * MM) + r * MAT;
            for (int j = 0; j < MAT; ++j) o[j] = R[j];
        }
    }
}

extern "C" void kernel_launch(void* const* d_in, const int* in_sizes, int n_in,
                              void* d_out, int out_size, void* d_ws, size_t ws_size,
                              hipStream_t stream) {
    const float* z     = (const float*)d_in[0];   // [4096, 16, 512] f32
    const float* basis = (const float*)d_in[1];   // [16, 512, 10, 10] f32
    float* out = (float*)d_out;                   // [4096, 16, 10, 10] f32

    dim3 grid(B_TOT / 16, NUM_WS);                // 256 x 16 workgroups
    lie_expm_fused<<<grid, THREADS, 0, stream>>>(z, basis, out);
}